// AutoformerNetwork_27023934226998
// MI455X (gfx1250) — compile-verified
//
#include <hip/hip_runtime.h>
#include <hip/hip_bf16.h>

// ---------------- model constants ----------------
#define B_    16
#define L_    1536
#define D_    512
#define H_    8
#define DK_   64
#define PF_   2048
#define IND_  7
#define KTOP  40                      // FACTOR * ceil(log(1536)) = 5*8
#define ROWS  (B_*L_)                 // 24576
#define BHL   (B_*H_*L_)              // 196608
#define SZBL  ((size_t)B_*L_*D_)      // 12582912
#define CHUNK_M 6144
#define NCHUNK  4

typedef float v2f __attribute__((ext_vector_type(2)));
typedef float v8f __attribute__((ext_vector_type(8)));

// ================= kernel 1: effective projection weights =================
// qscalar[b,h,l] = x[b,l,:] . Weff[h] + beff[h], where
// Weff[h][c] = sum_d W[c, h*64+d]*w1[d];  beff[h] = sum_d b[h*64+d]*w1[d] + b1.
// eff layout: [proj(3)][h(8)][c(512)] then biases [proj(3)][h(8)] at 12288.
__global__ void k_eff(const float* __restrict__ Wq, const float* __restrict__ bq,
                      const float* __restrict__ wq1, const float* __restrict__ bq1,
                      const float* __restrict__ Wk, const float* __restrict__ bk,
                      const float* __restrict__ wk1, const float* __restrict__ bk1,
                      const float* __restrict__ Wv, const float* __restrict__ bv,
                      const float* __restrict__ wv1, const float* __restrict__ bv1,
                      float* __restrict__ eff)
{
    int gid = blockIdx.x * blockDim.x + threadIdx.x;
    if (gid < 3*H_*D_) {
        int proj = gid / (H_*D_);
        int rem  = gid % (H_*D_);
        int h = rem / D_;
        int c = rem % D_;
        const float* W  = (proj==0) ? Wq  : (proj==1) ? Wk  : Wv;
        const float* w1 = (proj==0) ? wq1 : (proj==1) ? wk1 : wv1;
        float acc = 0.f;
        #pragma unroll 8
        for (int d = 0; d < DK_; ++d) acc += W[c*D_ + h*DK_ + d] * w1[d];
        eff[gid] = acc;
    } else if (gid < 3*H_*D_ + 3*H_) {
        int o = gid - 3*H_*D_;
        int proj = o / H_, h = o % H_;
        const float* b  = (proj==0) ? bq  : (proj==1) ? bk  : bv;
        const float* w1 = (proj==0) ? wq1 : (proj==1) ? wk1 : wv1;
        const float* b1 = (proj==0) ? bq1 : (proj==1) ? bk1 : bv1;
        float acc = b1[0];
        for (int d = 0; d < DK_; ++d) acc += b[h*DK_ + d] * w1[d];
        eff[gid] = acc;
    }
}

// ================= kernel 2: q/k/v scalar projections =================
// One block handles 8 consecutive (b,l) rows. q from qin, k/v from kin.
__global__ __launch_bounds__(256) void k_qkv(
    const float* __restrict__ qin, const float* __restrict__ kin,
    const float* __restrict__ eff,
    float* __restrict__ qo, float* __restrict__ ko, float* __restrict__ vo)
{
    __shared__ float xq[8][D_];
    __shared__ float xk[8][D_];
    int t = threadIdx.x;
    int row0 = blockIdx.x * 8;
    #pragma unroll
    for (int i = 0; i < 16; ++i) {
        int idx = t + i * 256;           // 0..4095
        int r = idx >> 9;
        int c = idx & 511;
        xq[r][c] = qin[(size_t)(row0 + r) * D_ + c];
        xk[r][c] = kin[(size_t)(row0 + r) * D_ + c];
    }
    __syncthreads();
    if (t < 192) {
        int r = t / 24, o = t % 24;
        int proj = o / 8, h = o % 8;
        const float* src = (proj == 0) ? xq[r] : xk[r];
        const float* w = eff + (size_t)proj * H_ * D_ + (size_t)h * D_;
        float acc = eff[3*H_*D_ + proj*H_ + h];
        for (int c = 0; c < D_; ++c) acc += src[c] * w[c];
        int row = row0 + r;
        int b = row / L_, l = row % L_;
        float* out = (proj == 0) ? qo : (proj == 1) ? ko : vo;
        out[((size_t)b * H_ + h) * L_ + l] = acc;
    }
}

// ========== kernel 3: fused autocorrelation score + top-40 + softmax + gather ==========
// One block per (b,h). score[j] = (1/L) sum_{t>=j} q[t]*k[t-j]  (linear corr == FFT path)
// attn[l] = sum_k w_k * v[(l + idx_k) % L]
__global__ __launch_bounds__(256) void k_autocorr(
    const float* __restrict__ q, const float* __restrict__ k,
    const float* __restrict__ v, float* __restrict__ attn)
{
    __shared__ float sq[L_];
    __shared__ float sk[L_];
    __shared__ float sc[L_];
    __shared__ float rv[256];
    __shared__ int   ri[256];
    __shared__ float wl[KTOP];
    __shared__ int   il[KTOP];
    int t = threadIdx.x;
    size_t base = (size_t)blockIdx.x * L_;
    for (int i = t; i < L_; i += 256) { sq[i] = q[base + i]; sk[i] = k[base + i]; }
    __syncthreads();
    for (int j = t; j < L_; j += 256) {
        float acc = 0.f;
        for (int tt = j; tt < L_; ++tt) acc += sq[tt] * sk[tt - j];
        sc[j] = acc * (1.0f / (float)L_);
    }
    __syncthreads();
    // iterative top-40 (argmax selection; ties -> lower index like lax.top_k)
    for (int it = 0; it < KTOP; ++it) {
        float bv = -1e30f; int bi = 0;
        for (int j = t; j < L_; j += 256) {
            float vv = sc[j];
            if (vv > bv) { bv = vv; bi = j; }
        }
        rv[t] = bv; ri[t] = bi;
        __syncthreads();
        for (int s = 128; s > 0; s >>= 1) {
            if (t < s) {
                if (rv[t+s] > rv[t] || (rv[t+s] == rv[t] && ri[t+s] < ri[t])) {
                    rv[t] = rv[t+s]; ri[t] = ri[t+s];
                }
            }
            __syncthreads();
        }
        if (t == 0) { wl[it] = rv[0]; il[it] = ri[0]; sc[ri[0]] = -1e38f; }
        __syncthreads();
    }
    if (t == 0) {   // softmax over 40 weights
        float m = wl[0];
        for (int i = 1; i < KTOP; ++i) m = fmaxf(m, wl[i]);
        float s = 0.f;
        for (int i = 0; i < KTOP; ++i) { wl[i] = expf(wl[i] - m); s += wl[i]; }
        float inv = 1.f / s;
        for (int i = 0; i < KTOP; ++i) wl[i] *= inv;
    }
    for (int i = t; i < L_; i += 256) sq[i] = v[base + i];   // reuse sq for v
    __syncthreads();
    for (int l = t; l < L_; l += 256) {
        float acc = 0.f;
        #pragma unroll 8
        for (int kk = 0; kk < KTOP; ++kk) {
            int p = l + il[kk]; if (p >= L_) p -= L_;
            acc += wl[kk] * sq[p];
        }
        attn[base + l] = acc;
    }
}

// ========== kernel 4: output projection + residual + channel-decomp ==========
// NOTE reference does out.reshape(B,L,H) on a (B,H,L) array -> raw reinterpret:
// the H values for row (b,l) are attn_flat[b, l*8 .. l*8+7].
__global__ __launch_bounds__(256) void k_outproj_decomp(
    const float* __restrict__ attn, const float* __restrict__ Wo,
    const float* __restrict__ bo, const float* __restrict__ res,
    float* __restrict__ s_out, float* __restrict__ t_out)
{
    __shared__ float ah[H_];
    __shared__ float yr[D_];
    int t = threadIdx.x;
    int row = blockIdx.x;
    int b = row / L_, l = row % L_;
    if (t < H_) ah[t] = attn[(size_t)b * H_ * L_ + (size_t)l * H_ + t];
    __syncthreads();
    for (int c = t; c < D_; c += 256) {
        float acc = bo[c] + res[(size_t)row * D_ + c];
        #pragma unroll
        for (int h = 0; h < H_; ++h) acc += ah[h] * Wo[h * D_ + c];
        yr[c] = acc;
    }
    __syncthreads();
    for (int c = t; c < D_; c += 256) {
        float left  = (c > 0)      ? yr[c-1] : 0.f;
        float right = (c < D_ - 1) ? yr[c+1] : 0.f;
        float tr = (left + yr[c] + right) * (1.f / 3.f);
        s_out[(size_t)row * D_ + c] = yr[c] - tr;
        if (t_out) t_out[(size_t)row * D_ + c] = tr;
    }
}

// ================= kernel 5: decomp only =================
__global__ __launch_bounds__(256) void k_decomp(
    const float* __restrict__ y, float* __restrict__ s_out, float* __restrict__ t_out)
{
    __shared__ float yr[D_];
    int t = threadIdx.x;
    int row = blockIdx.x;
    for (int c = t; c < D_; c += 256) yr[c] = y[(size_t)row * D_ + c];
    __syncthreads();
    for (int c = t; c < D_; c += 256) {
        float left  = (c > 0)      ? yr[c-1] : 0.f;
        float right = (c < D_ - 1) ? yr[c+1] : 0.f;
        float tr = (left + yr[c] + right) * (1.f / 3.f);
        s_out[(size_t)row * D_ + c] = yr[c] - tr;
        if (t_out) t_out[(size_t)row * D_ + c] = tr;
    }
}

// ================= kernel 6: fp32 WMMA GEMM =================
// Out[M,N] = op(A[M,K] @ B[K,N] + bias[N]) (+Res[M,N]);  op = ReLU if relu!=0.
// Block tile 64x128, 8 waves of 32x32, K-staged through LDS in steps of 32,
// inner product via V_WMMA_F32_16X16X4_F32.
#define BM 64
#define BN 128
#define KT 32
__global__ __launch_bounds__(256) void k_gemm(
    const float* __restrict__ A, const float* __restrict__ Bm,
    const float* __restrict__ bias, const float* __restrict__ Res,
    float* __restrict__ Out, int M, int N, int K, int relu)
{
    __shared__ float As[BM][KT + 1];
    __shared__ float Bs[KT][BN + 1];
    int t = threadIdx.x;
    int lane = t & 31, wave = t >> 5;
    int lhalf = lane >> 4;         // 0/1
    int lmod  = lane & 15;
    int wm = (wave >> 2) * 32;     // 0,32
    int wn = (wave & 3) * 32;      // 0,32,64,96
    int m0 = blockIdx.y * BM;
    int n0 = blockIdx.x * BN;

    v8f acc[2][2];
    #pragma unroll
    for (int i = 0; i < 2; ++i)
        #pragma unroll
        for (int j = 0; j < 2; ++j)
            #pragma unroll
            for (int e = 0; e < 8; ++e) acc[i][j][e] = 0.f;

    for (int k0 = 0; k0 < K; k0 += KT) {
        // stage A: 64x32 floats = 512 float4, 2 per thread
        #pragma unroll
        for (int i = 0; i < 2; ++i) {
            int idx = t + i * 256;
            int r  = idx >> 3;
            int c4 = (idx & 7) << 2;
            float4 va = *reinterpret_cast<const float4*>(A + (size_t)(m0 + r) * K + k0 + c4);
            As[r][c4+0] = va.x; As[r][c4+1] = va.y; As[r][c4+2] = va.z; As[r][c4+3] = va.w;
        }
        // stage B: 32x128 floats = 1024 float4, 4 per thread
        #pragma unroll
        for (int i = 0; i < 4; ++i) {
            int idx = t + i * 256;
            int r  = idx >> 5;
            int c4 = (idx & 31) << 2;
            float4 vb = *reinterpret_cast<const float4*>(Bm + (size_t)(k0 + r) * N + n0 + c4);
            Bs[r][c4+0] = vb.x; Bs[r][c4+1] = vb.y; Bs[r][c4+2] = vb.z; Bs[r][c4+3] = vb.w;
        }
        __syncthreads();
        for (int kk = 0; kk < KT; kk += 4) {
            v2f af[2], bf[2];
            #pragma unroll
            for (int i = 0; i < 2; ++i) {
                // A 16x4 frag: lane l, reg r -> A[M = l%16, K = (l/16)*2 + r]
                int rrow = wm + i * 16 + lmod;
                int rcol = kk + lhalf * 2;
                af[i][0] = As[rrow][rcol];
                af[i][1] = As[rrow][rcol + 1];
            }
            #pragma unroll
            for (int j = 0; j < 2; ++j) {
                // B 4x16 frag: lane l, reg r -> B[K = (l/16)*2 + r, N = l%16]
                int bcol = wn + j * 16 + lmod;
                int brow = kk + lhalf * 2;
                bf[j][0] = Bs[brow][bcol];
                bf[j][1] = Bs[brow + 1][bcol];
            }
            #pragma unroll
            for (int i = 0; i < 2; ++i)
                #pragma unroll
                for (int j = 0; j < 2; ++j)
                    acc[i][j] = __builtin_amdgcn_wmma_f32_16x16x4_f32(
                        false, af[i], false, bf[j], (short)0, acc[i][j], false, false);
        }
        __syncthreads();
    }
    // epilogue: C layout lane l, reg r -> (M = (l/16)*8 + r, N = l%16)
    #pragma unroll
    for (int i = 0; i < 2; ++i)
        #pragma unroll
        for (int j = 0; j < 2; ++j) {
            int nn = n0 + wn + j * 16 + lmod;
            float bvv = bias ? bias[nn] : 0.f;
            #pragma unroll
            for (int r = 0; r < 8; ++r) {
                int mm = m0 + wm + i * 16 + lhalf * 8 + r;
                float vv = acc[i][j][r] + bvv;
                if (relu) vv = fmaxf(vv, 0.f);
                if (Res) vv += Res[(size_t)mm * N + nn];
                Out[(size_t)mm * N + nn] = vv;
            }
        }
}

// ================= kernel 7: trend projections =================
__global__ __launch_bounds__(64) void k_trend(
    const float* __restrict__ dec_trend,
    const float* __restrict__ T1, const float* __restrict__ T2, const float* __restrict__ T3,
    const float* __restrict__ Wp1, const float* __restrict__ bp1,
    const float* __restrict__ Wp2, const float* __restrict__ bp2,
    const float* __restrict__ Wp3, const float* __restrict__ bp3,
    float* __restrict__ out)
{
    __shared__ float r1[D_], r2[D_], r3[D_];
    int t = threadIdx.x;
    int row = blockIdx.x;
    for (int c = t; c < D_; c += 64) {
        r1[c] = T1[(size_t)row*D_ + c];
        r2[c] = T2[(size_t)row*D_ + c];
        r3[c] = T3[(size_t)row*D_ + c];
    }
    __syncthreads();
    if (t < IND_) {
        float acc = dec_trend[(size_t)row*IND_ + t] + bp1[t] + bp2[t] + bp3[t];
        for (int c = 0; c < D_; ++c)
            acc += r1[c]*Wp1[c*IND_ + t] + r2[c]*Wp2[c*IND_ + t] + r3[c]*Wp3[c*IND_ + t];
        out[(size_t)row*IND_ + t] = acc;
    }
}

// ================= host orchestration =================
struct AttnP {
    const float *Wq,*bq,*Wk,*bk,*Wv,*bv,*wq1,*bq1,*wk1,*bk1,*wv1,*bv1,*Wo,*bo;
};
static AttnP attn_at(void* const* d_in, int base) {
    AttnP p;
    p.Wq = (const float*)d_in[base+0];  p.bq  = (const float*)d_in[base+1];
    p.Wk = (const float*)d_in[base+2];  p.bk  = (const float*)d_in[base+3];
    p.Wv = (const float*)d_in[base+4];  p.bv  = (const float*)d_in[base+5];
    p.wq1= (const float*)d_in[base+6];  p.bq1 = (const float*)d_in[base+7];
    p.wk1= (const float*)d_in[base+8];  p.bk1 = (const float*)d_in[base+9];
    p.wv1= (const float*)d_in[base+10]; p.bv1 = (const float*)d_in[base+11];
    p.Wo = (const float*)d_in[base+12]; p.bo  = (const float*)d_in[base+13];
    return p;
}
struct FfnP { const float *W1,*b1,*W2,*b2; };
static FfnP ffn_at(void* const* d_in, int base) {
    FfnP f;
    f.W1 = (const float*)d_in[base+0]; f.b1 = (const float*)d_in[base+1];
    f.W2 = (const float*)d_in[base+2]; f.b2 = (const float*)d_in[base+3];
    return f;
}

static void run_attn(const AttnP& p, const float* qin, const float* kin,
                     const float* res, float* s_out, float* t_out,
                     float* qb, float* kb, float* vb, float* ab, float* eff,
                     hipStream_t stream)
{
    k_eff<<<49, 256, 0, stream>>>(p.Wq,p.bq,p.wq1,p.bq1,
                                  p.Wk,p.bk,p.wk1,p.bk1,
                                  p.Wv,p.bv,p.wv1,p.bv1, eff);
    k_qkv<<<ROWS/8, 256, 0, stream>>>(qin, kin, eff, qb, kb, vb);
    k_autocorr<<<B_*H_, 256, 0, stream>>>(qb, kb, vb, ab);
    k_outproj_decomp<<<ROWS, 256, 0, stream>>>(ab, p.Wo, p.bo, res, s_out, t_out);
}

static void run_ffn(const float* S, const FfnP& f, float* Hb, float* Y, hipStream_t stream)
{
    for (int ch = 0; ch < NCHUNK; ++ch) {
        const float* Ac = S + (size_t)ch * CHUNK_M * D_;
        float* Yc = Y + (size_t)ch * CHUNK_M * D_;
        dim3 g1(PF_/BN, CHUNK_M/BM);
        k_gemm<<<g1, 256, 0, stream>>>(Ac, f.W1, f.b1, nullptr, Hb, CHUNK_M, PF_, D_, 1);
        dim3 g2(D_/BN, CHUNK_M/BM);
        k_gemm<<<g2, 256, 0, stream>>>(Hb, f.W2, f.b2, Ac, Yc, CHUNK_M, D_, PF_, 0);
    }
}

extern "C" void kernel_launch(void* const* d_in, const int* in_sizes, int n_in,
                              void* d_out, int out_size, void* d_ws, size_t ws_size,
                              hipStream_t stream)
{
    (void)in_sizes; (void)n_in; (void)out_size; (void)ws_size;
    const float* dec_season = (const float*)d_in[0];
    const float* dec_trend  = (const float*)d_in[1];
    const float* enc_input  = (const float*)d_in[2];

    float* ws = (float*)d_ws;
    float* X  = ws;                 // activations / S2 / encoder output
    float* S1 = ws + 1*SZBL;
    float* Y  = ws + 2*SZBL;
    float* T1 = ws + 3*SZBL;
    float* T2 = ws + 4*SZBL;
    float* T3 = ws + 5*SZBL;
    float* Hb = ws + 6*SZBL;        // FFN hidden chunk (6144 x 2048)
    float* qb = ws + 7*SZBL;
    float* kb = qb + BHL;
    float* vb = kb + BHL;
    float* ab = vb + BHL;
    float* eff = ab + BHL;          // 12312 floats

    float* out_ds = (float*)d_out;
    float* out_dt = out_ds + SZBL;

    // ---------------- encoder (2 layers) ----------------
    const float* x = enc_input;
    for (int i = 0; i < 2; ++i) {
        int base = 3 + i * 18;
        AttnP ap = attn_at(d_in, base);
        FfnP  fp = ffn_at(d_in, base + 14);
        run_attn(ap, x, x, x, S1, nullptr, qb, kb, vb, ab, eff, stream);
        run_ffn(S1, fp, Hb, Y, stream);
        k_decomp<<<ROWS, 256, 0, stream>>>(Y, X, nullptr);
        x = X;
    }

    // ---------------- decoder (1 layer) ----------------
    int db = 3 + 2 * 18;                       // 39
    AttnP sp = attn_at(d_in, db);              // self
    AttnP cp = attn_at(d_in, db + 14);         // cross
    FfnP  dfp = ffn_at(d_in, db + 28);
    const float* Wp1 = (const float*)d_in[db + 32];
    const float* bp1 = (const float*)d_in[db + 33];
    const float* Wp2 = (const float*)d_in[db + 34];
    const float* bp2 = (const float*)d_in[db + 35];
    const float* Wp3 = (const float*)d_in[db + 36];
    const float* bp3 = (const float*)d_in[db + 37];

    // self-attention: s1,t1 = decomp(attn(ds)+ds)
    run_attn(sp, dec_season, dec_season, dec_season, S1, T1, qb, kb, vb, ab, eff, stream);
    // cross-attention: s2,t2 = decomp(attn(s1, x_enc)+s1); S2 overwrites X (enc out
    // is consumed by k_qkv before k_outproj_decomp writes X — serialized on stream)
    run_attn(cp, S1, X, S1, X, T2, qb, kb, vb, ab, eff, stream);
    // FFN: s3,t3 = decomp(ffn(s2)+s2); season output straight into d_out
    run_ffn(X, dfp, Hb, Y, stream);
    k_decomp<<<ROWS, 256, 0, stream>>>(Y, out_ds, T3);
    // trend accumulation
    k_trend<<<ROWS, 64, 0, stream>>>(dec_trend, T1, T2, T3,
                                     Wp1, bp1, Wp2, bp2, Wp3, bp3, out_dt);
}